// EdgeOrient_54803782697130
// MI455X (gfx1250) — compile-verified
//
#include <hip/hip_runtime.h>
#include <hip/hip_bf16.h>

// ---------------------------------------------------------------------------
// EdgeOrient GNN on MI455X (gfx1250, wave32)
//   per layer:  m_up = scatter_sum(h[src]*orient)  (L2-resident atomics)
//               m_dn = scatter_sum(h[src]*orient)
//               h    = tanh(h @ Ws^T + m_up @ Wu^T + m_dn @ Wd^T)   [fp32 WMMA]
//   readout:    pooled = segsum(|h|, batch);  relu-linear;  linear
// ---------------------------------------------------------------------------

typedef __attribute__((ext_vector_type(2))) float v2f;
typedef __attribute__((ext_vector_type(8))) float v8f;

#define HDIM 64

// ---------------------------------------------------------------- zero fill
__global__ void eo_zero_kernel(float4* __restrict__ p, long n4) {
    long i      = (long)blockIdx.x * blockDim.x + threadIdx.x;
    long stride = (long)gridDim.x * blockDim.x;
    for (; i < n4; i += stride) p[i] = make_float4(0.f, 0.f, 0.f, 0.f);
}

// ------------------------------------------------- gather * orient -> scatter
// one wave32 per adjacency entry; lane handles float2 of the 64-wide feature row
__global__ void eo_agg_kernel(const float* __restrict__ h,
                              const int*   __restrict__ src,
                              const int*   __restrict__ dst,
                              const float* __restrict__ orient,
                              float*                     m,
                              int nadj) {
    int e    = (int)((blockIdx.x * (unsigned)blockDim.x + threadIdx.x) >> 5);
    int lane = threadIdx.x & 31;
    if (e >= nadj) return;
    // per-wave uniform scalars -> SGPRs
    int   s = __builtin_amdgcn_readfirstlane(src[e]);
    int   d = __builtin_amdgcn_readfirstlane(dst[e]);
    float o = orient[e];
    v2f v = *(const v2f*)(h + (size_t)s * HDIM + lane * 2);
    float* mp = m + (size_t)d * HDIM + lane * 2;
    atomicAdd(mp,     v.x * o);   // global_atomic_add_f32, non-returning
    atomicAdd(mp + 1, v.y * o);
}

// ---------------------------------------------------- fused triple GEMM+tanh
// block = one 16-row strip of the [E,64] matrices, 4 waves = 4 x 16-col tiles.
// out may alias mu: each strip's rows are consumed only by its own block, and
// __syncthreads() separates all A-fragment loads from the stores.
__global__ void eo_conv_kernel(const float* __restrict__ h,
                               const float*              mu,
                               const float* __restrict__ md,
                               const float* __restrict__ Ws,
                               const float* __restrict__ Wu,
                               const float* __restrict__ Wd,
                               float*                    out) {
    const int wave  = threadIdx.x >> 5;       // 0..3 -> output col tile
    const int lane  = threadIdx.x & 31;
    const int row0  = blockIdx.x * 16;
    const int col0  = wave * 16;
    const int r     = lane & 15;              // M (A) / N (B) index within tile
    const int khalf = (lane >> 4) * 2;        // lanes 16..31 hold K+2,K+3

    // A fragments: rows of h / m_up / m_dn   (contiguous float2 per lane)
    const float* ha = h  + (size_t)(row0 + r) * HDIM + khalf;
    const float* ua = mu + (size_t)(row0 + r) * HDIM + khalf;
    const float* da = md + (size_t)(row0 + r) * HDIM + khalf;
    // B fragments: B = W^T tile -> same contiguous pattern on rows of W
    const float* sb = Ws + (size_t)(col0 + r) * HDIM + khalf;
    const float* ub = Wu + (size_t)(col0 + r) * HDIM + khalf;
    const float* db = Wd + (size_t)(col0 + r) * HDIM + khalf;

    v8f c = {};
#pragma unroll
    for (int k = 0; k < HDIM; k += 4) {
        v2f a0 = *(const v2f*)(ha + k);
        v2f b0 = *(const v2f*)(sb + k);
        c = __builtin_amdgcn_wmma_f32_16x16x4_f32(false, a0, false, b0,
                                                  (short)0, c, false, false);
        v2f a1 = *(const v2f*)(ua + k);
        v2f b1 = *(const v2f*)(ub + k);
        c = __builtin_amdgcn_wmma_f32_16x16x4_f32(false, a1, false, b1,
                                                  (short)0, c, false, false);
        v2f a2 = *(const v2f*)(da + k);
        v2f b2 = *(const v2f*)(db + k);
        c = __builtin_amdgcn_wmma_f32_16x16x4_f32(false, a2, false, b2,
                                                  (short)0, c, false, false);
    }

    __syncthreads();   // all waves done reading this strip -> in-place store OK

    // D layout: lane<16 -> N=lane, M=v ; lane>=16 -> N=lane-16, M=v+8
    const int mbase = (lane >> 4) * 8;
    const int ccol  = col0 + (lane & 15);
#pragma unroll
    for (int v = 0; v < 8; ++v)
        out[(size_t)(row0 + mbase + v) * HDIM + ccol] = tanhf(c[v]);
}

// -------------------------------------- segment-sum(|h|) over sorted batch id
// one wave per 64-edge chunk; flush atomics only on segment boundary / chunk end
__global__ void eo_pool_kernel(const float* __restrict__ h,
                               const int*   __restrict__ batch,
                               float*                    pooled,
                               int E) {
    int wg   = (int)((blockIdx.x * (unsigned)blockDim.x + threadIdx.x) >> 5);
    int lane = threadIdx.x & 31;
    int e0 = wg * 64;
    if (e0 >= E) return;
    int e1 = e0 + 64; if (e1 > E) e1 = E;

    float ax = 0.f, ay = 0.f;
    int cur = __builtin_amdgcn_readfirstlane(batch[e0]);
    for (int e = e0; e < e1; ++e) {
        int b = __builtin_amdgcn_readfirstlane(batch[e]);
        if (b != cur) {
            float* pp = pooled + (size_t)cur * HDIM + lane * 2;
            atomicAdd(pp, ax); atomicAdd(pp + 1, ay);
            ax = 0.f; ay = 0.f; cur = b;
        }
        v2f v = *(const v2f*)(h + (size_t)e * HDIM + lane * 2);
        ax += fabsf(v.x); ay += fabsf(v.y);
    }
    float* pp = pooled + (size_t)cur * HDIM + lane * 2;
    atomicAdd(pp, ax); atomicAdd(pp + 1, ay);
}

// ------------------------------------------------------------- tiny readouts
__global__ void eo_lin1_kernel(const float* __restrict__ pooled,
                               const float* __restrict__ w,
                               const float* __restrict__ bias,
                               float* __restrict__ z) {
    int idx = blockIdx.x * blockDim.x + threadIdx.x;   // B*H = 4096
    int b = idx >> 6, j = idx & 63;
    const float* pr = pooled + b * HDIM;
    const float* wr = w + j * HDIM;
    float acc = bias[j];
#pragma unroll 8
    for (int k = 0; k < HDIM; ++k) acc += pr[k] * wr[k];
    z[idx] = acc > 0.f ? acc : 0.f;
}

__global__ void eo_lin2_kernel(const float* __restrict__ z,
                               const float* __restrict__ w,
                               const float* __restrict__ bias,
                               float* __restrict__ out,
                               int B, int C) {
    int idx = blockIdx.x * blockDim.x + threadIdx.x;   // B*C = 640
    if (idx >= B * C) return;
    int b = idx / C, cc = idx % C;
    const float* zr = z + b * HDIM;
    const float* wr = w + cc * HDIM;
    float acc = bias[cc];
#pragma unroll 8
    for (int k = 0; k < HDIM; ++k) acc += zr[k] * wr[k];
    out[idx] = acc;
}

// ---------------------------------------------------------------------------
extern "C" void kernel_launch(void* const* d_in, const int* in_sizes, int n_in,
                              void* d_out, int out_size, void* d_ws, size_t ws_size,
                              hipStream_t stream) {
    const float* x         = (const float*)d_in[0];
    const int*   up_index  = (const int*)  d_in[1];   // [2, NADJ]
    const float* up_orient = (const float*)d_in[2];
    const int*   dn_index  = (const int*)  d_in[3];
    const float* dn_orient = (const float*)d_in[4];
    const int*   batch     = (const int*)  d_in[5];
    const float* W_up      = (const float*)d_in[6];   // [L,H,H]
    const float* W_dn      = (const float*)d_in[7];
    const float* W_self    = (const float*)d_in[8];
    const float* lin1_w    = (const float*)d_in[9];
    const float* lin1_b    = (const float*)d_in[10];
    const float* lin2_w    = (const float*)d_in[11];
    const float* lin2_b    = (const float*)d_in[12];
    float* out = (float*)d_out;

    const int E    = in_sizes[0] / HDIM;   // 200000
    const int NADJ = in_sizes[2];          // 3200000
    const int B    = 64;
    const int C    = out_size / B;         // 10
    const int Lnum = 3;

    const size_t mat = (size_t)E * HDIM;   // floats per [E,H] buffer
    float* bufs[3];
    bufs[0] = (float*)d_ws;
    bufs[1] = bufs[0] + mat;
    bufs[2] = bufs[1] + mat;
    float* pooled = bufs[2] + mat;         // B*H
    float* z      = pooled + B * HDIM;     // B*H

    const int aggBlocks  = (NADJ * 32 + 255) / 256;       // 1 wave / entry
    const int convBlocks = E / 16;                        // E divisible by 16
    const int zeroBlocks = 2048;

    const float* hcur = x;
    for (int l = 0; l < Lnum; ++l) {
        float* mu = bufs[l % 3];
        float* md = bufs[(l + 1) % 3];
        eo_zero_kernel<<<zeroBlocks, 256, 0, stream>>>((float4*)mu, (long)(mat / 4));
        eo_zero_kernel<<<zeroBlocks, 256, 0, stream>>>((float4*)md, (long)(mat / 4));
        eo_agg_kernel<<<aggBlocks, 256, 0, stream>>>(hcur, up_index, up_index + NADJ,
                                                     up_orient, mu, NADJ);
        eo_agg_kernel<<<aggBlocks, 256, 0, stream>>>(hcur, dn_index, dn_index + NADJ,
                                                     dn_orient, md, NADJ);
        eo_conv_kernel<<<convBlocks, 128, 0, stream>>>(hcur, mu, md,
                                                       W_self + (size_t)l * HDIM * HDIM,
                                                       W_up   + (size_t)l * HDIM * HDIM,
                                                       W_dn   + (size_t)l * HDIM * HDIM,
                                                       mu /* in-place */);
        hcur = mu;
    }

    // readout
    eo_zero_kernel<<<1, 256, 0, stream>>>((float4*)pooled, (long)((2 * B * HDIM) / 4));
    int poolWaves  = (E + 63) / 64;
    int poolBlocks = (poolWaves + 7) / 8;
    eo_pool_kernel<<<poolBlocks, 256, 0, stream>>>(hcur, batch, pooled, E);
    eo_lin1_kernel<<<(B * HDIM) / 256, 256, 0, stream>>>(pooled, lin1_w, lin1_b, z);
    eo_lin2_kernel<<<(B * C + 255) / 256, 256, 0, stream>>>(z, lin2_w, lin2_b, out, B, C);
}